// refGATConv_52871047413956
// MI455X (gfx1250) — compile-verified
//
#include <hip/hip_runtime.h>
#include <hip/hip_bf16.h>

#define N_NODES 100000
#define N_EDGES 1600000
#define IN_CH   256
#define HC      128   // HEADS*OUT_CH
#define NEG_SLOPE 0.2f

typedef __attribute__((ext_vector_type(2))) float v2f;
typedef __attribute__((ext_vector_type(8))) float v8f;

// ---------------------------------------------------------------------------
// Kernel 1: feat[N,128] = x[N,256] @ W.T + b   using V_WMMA_F32_16X16X4_F32.
// Block = 256 threads = 8 waves; wave w computes output tile
// rows [blockIdx*16, +16) x cols [w*16, +16). K-loop: 64 fp32 WMMAs.
// A (16x4 f32) lane layout: lanes 0-15 -> M=lane, K={k,k+1}; lanes 16-31 ->
// M=lane-16, K={k+2,k+3}. B (4x16) mirrors with N in lane&15. (ISA 7.12.2)
// ---------------------------------------------------------------------------
__global__ __launch_bounds__(256) void gat_gemm_wmma(
    const float* __restrict__ x, const float* __restrict__ W,
    const float* __restrict__ bias, float* __restrict__ feat) {
  const int lane  = threadIdx.x & 31;
  const int wave  = threadIdx.x >> 5;       // column tile 0..7
  const int row0  = blockIdx.x * 16;        // 100000/16 = 6250 exact
  const int col0  = wave * 16;

  const int mr    = lane & 15;
  const int khalf = (lane >> 4) * 2;        // 0 or 2

  const float* xrow = x + (size_t)(row0 + mr) * IN_CH + khalf;  // A: row m
  const float* wrow = W + (size_t)(col0 + mr) * IN_CH + khalf;  // B[k][n] = W[n][k]

  v8f acc = {};
#pragma unroll 8
  for (int k = 0; k < IN_CH; k += 4) {
    v2f a = *(const v2f*)(xrow + k);
    v2f b = *(const v2f*)(wrow + k);
    acc = __builtin_amdgcn_wmma_f32_16x16x4_f32(
        /*neg_a=*/false, a, /*neg_b=*/false, b,
        /*c_mod=*/(short)0, acc, /*reuse_a=*/false, /*reuse_b=*/false);
  }

  // D layout: lane n = lane&15 ; VGPR v -> M = v + (lane>=16 ? 8 : 0)
  const int nc  = lane & 15;
  const int mhi = (lane >> 4) * 8;
  const float bb = bias[col0 + nc];
#pragma unroll
  for (int v = 0; v < 8; ++v) {
    const int m = row0 + mhi + v;
    feat[(size_t)m * HC + col0 + nc] = acc[v] + bb;
  }
}

// ---------------------------------------------------------------------------
// Kernel 2: per-node attention scalars. One wave per node; each lane handles
// 4 channels (float4), wave-reduce with shfl. e_sum initialized to e_self.
// ---------------------------------------------------------------------------
__global__ __launch_bounds__(256) void gat_node_att(
    const float* __restrict__ feat, const float* __restrict__ att,
    float* __restrict__ e_src, float* __restrict__ e_tar,
    float* __restrict__ e_self, float* __restrict__ e_sum) {
  const int wave = threadIdx.x >> 5;
  const int lane = threadIdx.x & 31;
  const int n = blockIdx.x * 8 + wave;
  if (n >= N_NODES) return;

  const float4 f = *(const float4*)(feat + (size_t)n * HC + lane * 4);
  const float* ap = att + lane * 8;  // att is (128,2) row-major
  float s0 = f.x * ap[0] + f.y * ap[2] + f.z * ap[4] + f.w * ap[6];
  float s1 = f.x * ap[1] + f.y * ap[3] + f.z * ap[5] + f.w * ap[7];
#pragma unroll
  for (int off = 16; off > 0; off >>= 1) {
    s0 += __shfl_down(s0, off);
    s1 += __shfl_down(s1, off);
  }
  if (lane == 0) {
    const float v  = s0 + s1;
    const float lr = v > 0.f ? v : NEG_SLOPE * v;
    const float es = __expf(lr);
    e_src[n]  = s0;
    e_tar[n]  = s1;
    e_self[n] = es;
    e_sum[n]  = es;   // init segment sum with self term
  }
}

// ---------------------------------------------------------------------------
// Kernel 3: per-edge weight + atomic segment-sum into e_sum[tar].
// ---------------------------------------------------------------------------
__global__ __launch_bounds__(256) void gat_edge_w(
    const long long* __restrict__ ei, const float* __restrict__ e_src,
    const float* __restrict__ e_tar, float* __restrict__ ew,
    float* __restrict__ e_sum) {
  const int e = blockIdx.x * blockDim.x + threadIdx.x;
  if (e >= N_EDGES) return;
  const int t = (int)ei[e];
  const int s = (int)ei[N_EDGES + e];
  const float v  = e_src[s] + e_tar[t];
  const float lr = v > 0.f ? v : NEG_SLOPE * v;
  const float w  = __expf(lr);
  ew[e] = w;
  __hip_atomic_fetch_add(e_sum + t, w, __ATOMIC_RELAXED, __HIP_MEMORY_SCOPE_AGENT);
}

// ---------------------------------------------------------------------------
// Kernel 4: self-loop term; fully overwrites out (clears harness poison).
// Thread per float4 of out.
// ---------------------------------------------------------------------------
__global__ __launch_bounds__(256) void gat_self_out(
    const float* __restrict__ feat, const float* __restrict__ e_self,
    const float* __restrict__ e_sum, float* __restrict__ out) {
  const int idx = blockIdx.x * blockDim.x + threadIdx.x;  // 32 float4 / node
  const int n = idx >> 5;
  if (n >= N_NODES) return;
  const int c = (idx & 31) * 4;
  const float a = e_self[n] / e_sum[n];
  float4 f = *(const float4*)(feat + (size_t)n * HC + c);
  f.x *= a; f.y *= a; f.z *= a; f.w *= a;
  *(float4*)(out + (size_t)n * HC + c) = f;
}

// ---------------------------------------------------------------------------
// Kernel 5: edge scatter. One wave per edge -> tar/src wave-uniform (SALU).
// Lane covers 4 channels: float4 read of feat[src] (L2-resident, 51 MB),
// 4x global_atomic_add_f32 into out[tar] (also L2-resident).
// ---------------------------------------------------------------------------
__global__ __launch_bounds__(256) void gat_edge_scatter(
    const long long* __restrict__ ei, const float* __restrict__ feat,
    const float* __restrict__ ew, const float* __restrict__ e_sum,
    float* __restrict__ out) {
  const int e = blockIdx.x * 8 + (threadIdx.x >> 5);
  if (e >= N_EDGES) return;
  const int lane = threadIdx.x & 31;
  const int t = (int)ei[e];
  const int s = (int)ei[N_EDGES + e];
  const float alpha = ew[e] / e_sum[t];
  const float4 f = *(const float4*)(feat + (size_t)s * HC + lane * 4);
  float* op = out + (size_t)t * HC + lane * 4;
  __hip_atomic_fetch_add(op + 0, f.x * alpha, __ATOMIC_RELAXED, __HIP_MEMORY_SCOPE_AGENT);
  __hip_atomic_fetch_add(op + 1, f.y * alpha, __ATOMIC_RELAXED, __HIP_MEMORY_SCOPE_AGENT);
  __hip_atomic_fetch_add(op + 2, f.z * alpha, __ATOMIC_RELAXED, __HIP_MEMORY_SCOPE_AGENT);
  __hip_atomic_fetch_add(op + 3, f.w * alpha, __ATOMIC_RELAXED, __HIP_MEMORY_SCOPE_AGENT);
}

// ---------------------------------------------------------------------------
extern "C" void kernel_launch(void* const* d_in, const int* in_sizes, int n_in,
                              void* d_out, int out_size, void* d_ws, size_t ws_size,
                              hipStream_t stream) {
  const float*     x   = (const float*)d_in[0];
  const long long* ei  = (const long long*)d_in[1];  // int64 (2, E)
  const float*     W   = (const float*)d_in[2];
  const float*     b   = (const float*)d_in[3];
  const float*     att = (const float*)d_in[4];
  float* out = (float*)d_out;

  // Workspace layout (256B aligned chunks): ~59 MB total
  char* ws = (char*)d_ws;
  float* feat   = (float*)ws;                       ws += ((size_t)N_NODES * HC * 4 + 255) & ~255ull;
  float* e_src  = (float*)ws;                       ws += ((size_t)N_NODES * 4 + 255) & ~255ull;
  float* e_tar  = (float*)ws;                       ws += ((size_t)N_NODES * 4 + 255) & ~255ull;
  float* e_self = (float*)ws;                       ws += ((size_t)N_NODES * 4 + 255) & ~255ull;
  float* e_sum  = (float*)ws;                       ws += ((size_t)N_NODES * 4 + 255) & ~255ull;
  float* ew     = (float*)ws;                       ws += ((size_t)N_EDGES * 4 + 255) & ~255ull;

  // 1) GEMM: 6250 row-tiles of 16; 8 waves cover the 128 output columns.
  gat_gemm_wmma<<<N_NODES / 16, 256, 0, stream>>>(x, W, b, feat);

  // 2) Node attention: 8 nodes per block.
  gat_node_att<<<(N_NODES + 7) / 8, 256, 0, stream>>>(feat, att, e_src, e_tar,
                                                      e_self, e_sum);

  // 3) Edge weights + segment sum (1.6M threads).
  gat_edge_w<<<(N_EDGES + 255) / 256, 256, 0, stream>>>(ei, e_src, e_tar, ew, e_sum);

  // 4) Self term (overwrites entire out).
  gat_self_out<<<(N_NODES * 32 + 255) / 256, 256, 0, stream>>>(feat, e_self, e_sum, out);

  // 5) Edge scatter: 8 edges (waves) per block.
  gat_edge_scatter<<<(N_EDGES + 7) / 8, 256, 0, stream>>>(ei, feat, ew, e_sum, out);
}